// CrossFusionModule_54485955117256
// MI455X (gfx1250) — compile-verified
//
#include <hip/hip_runtime.h>
#include <math.h>

// Problem constants (from the reference)
#define DIMD 256
#define BATCH 8
#define SEQ   2048
#define EPSV  1e-5f
#define LOG2E 1.4426950408889634f

typedef __attribute__((ext_vector_type(2))) float v2f;  // 16x4 f32 A / 4x16 f32 B per lane
typedef __attribute__((ext_vector_type(8))) float v8f;  // 16x16 f32 C/D per lane

// Workspace float layout:
//  [0]=alpha [1]=beta_c [2]=gamma_c [3]=delta [4]=c6 [5]=c7 [6]=c8 [7]=ln_gamma [8]=ln_beta
//  [16 ..)                 : A_t array  (BATCH*SEQ floats)
//  [16+BATCH*SEQ ..)       : V_s array  (BATCH*SEQ floats)
#define WS_A_OFF 16
#define WS_V_OFF (16 + BATCH * SEQ)

// ---------------------------------------------------------------------------
// Kernel 1: fuse all parameter-only contractions into 7 scalars.
// ---------------------------------------------------------------------------
__global__ __launch_bounds__(256)
void cf_scalars(const float* __restrict__ Wa, const float* __restrict__ ba,
                const float* __restrict__ Wv, const float* __restrict__ bv,
                const float* __restrict__ corr, const float* __restrict__ Wb,
                const float* __restrict__ bb, const float* __restrict__ gam,
                const float* __restrict__ bet, float* __restrict__ ws)
{
    __shared__ float red[256];
    const int e = threadIdx.x;

    float p = 0.f, q = 0.f;
    for (int d = 0; d < DIMD; ++d) {
        const float c = corr[d * DIMD + e];
        p = fmaf(Wa[d], c, p);
        q = fmaf(ba[d], c, q);
    }

    float part[7];
    part[0] = p * Wv[e];
    part[1] = p * bv[e];
    part[2] = q * Wv[e];
    part[3] = q * bv[e];
    part[4] = Wb[e] * Wa[e];
    part[5] = Wb[DIMD + e] * Wv[e];
    part[6] = 2.f * (Wb[e] * ba[e] + Wb[DIMD + e] * bv[e]);

    for (int k = 0; k < 7; ++k) {
        red[e] = part[k];
        __syncthreads();
        for (int off = 128; off > 0; off >>= 1) {
            if (e < off) red[e] += red[e + off];
            __syncthreads();
        }
        if (e == 0) ws[k] = red[0] + (k == 6 ? bb[0] : 0.f);
        __syncthreads();
    }
    if (e == 0) { ws[7] = gam[0]; ws[8] = bet[0]; }
}

// ---------------------------------------------------------------------------
// Kernel 2: flash-style online softmax over exact rank-2 cc tiles.
// One wave owns 16 query columns; loops 128 key tiles of 16.
// cc 16x16 tile via V_WMMA_F32_16X16X4_F32 (coefficients pre-scaled by log2e
// so v_exp_f32 needs no per-element multiply):
//   A[m,:] = [x_key[s0+m], 1, 0, 0]   (M = key axis)
//   B[:,n] = [r0(q_n), r1(q_n), 0, 0] (N = query axis)
// Lane halves see disjoint key rows (0-7 vs 8-15), so each half keeps an
// independent online-softmax state; halves are merged ONCE at the end
// (no per-tile ds_bpermute / dscnt waits).
// dir=0: queries = visual, keys = audio  -> A_t  (softmax over s)
// dir=1: queries = audio,  keys = visual -> V_s  (softmax over t)
// ---------------------------------------------------------------------------
__global__ __launch_bounds__(256)
void cf_attention(const float* __restrict__ xa, const float* __restrict__ xv,
                  float* __restrict__ ws)
{
    const float alpha = ws[0], betac = ws[1], gammac = ws[2], delta = ws[3];

    int idx = blockIdx.x;
    const int dir = idx & 1; idx >>= 1;
    const int b  = idx >> 4;
    const int qc = idx & 15;

    const int wave = threadIdx.x >> 5;
    const int lane = threadIdx.x & 31;
    const int n    = lane & 15;        // column (query) owned by this lane
    const bool hi  = lane >= 16;       // lane half -> K rows {2,3} of A/B (zero-padded)
    const int q0   = qc * 128 + wave * 16;

    const float* __restrict__ Xq = dir ? (xa + b * SEQ) : (xv + b * SEQ);
    const float* __restrict__ Xk = dir ? (xv + b * SEQ) : (xa + b * SEQ);
    float* __restrict__ outp = ws + (dir ? WS_V_OFF : WS_A_OFF) + b * SEQ + q0;

    // B matrix (constant over the key loop): exact cc coefficients per query,
    // pre-scaled by log2(e) -> WMMA emits logits directly in exp2 domain.
    const float xq = Xq[q0 + n];
    const float r0 = fmaf(alpha, xq, dir ? gammac : betac) * LOG2E; // * x_key
    const float r1 = fmaf(dir ? betac : gammac, xq, delta) * LOG2E; // * 1
    v2f Bm;
    Bm.x = hi ? 0.f : r0;   // K=0 row (lanes<16), K=2 row = 0 (lanes>=16)
    Bm.y = hi ? 0.f : r1;   // K=1 row          , K=3 row = 0

    // Per-half-lane online softmax state (disjoint key subsets per half).
    float rm = -3.0e38f, Z = 0.f, W = 0.f;

    #pragma unroll 2
    for (int s0 = 0; s0 < SEQ; s0 += 16) {
        // A matrix: key tile
        const float xk_lane = Xk[s0 + n];          // in-bounds for all lanes
        v2f Am;
        Am.x = hi ? 0.f : xk_lane;                 // K=0: x_key[s0+M]
        Am.y = hi ? 0.f : 1.0f;                    // K=1: ones column

        v8f acc = {};                              // C = 0
        acc = __builtin_amdgcn_wmma_f32_16x16x4_f32(
            false, Am, false, Bm, (short)0, acc, false, false);

        // This lane holds rows M = m + (hi?8:0) of its column; matching keys:
        const float* __restrict__ xk8 = Xk + s0 + (hi ? 8 : 0);

        // local max over this lane's 8 logits
        float tmax = fmaxf(acc[0], acc[1]);
        #pragma unroll
        for (int m = 2; m < 8; ++m) tmax = fmaxf(tmax, acc[m]);

        // online rescale of running state, then accumulate tile directly
        const float nm    = fmaxf(rm, tmax);
        const float s_old = __builtin_amdgcn_exp2f(rm - nm);
        Z *= s_old;
        W *= s_old;
        #pragma unroll
        for (int m = 0; m < 8; ++m) {
            const float ev = __builtin_amdgcn_exp2f(acc[m] - nm);
            Z += ev;
            W = fmaf(ev, xk8[m], W);
        }
        rm = nm;
    }

    // single cross-half merge (3 shuffles per wave total)
    const float om = __shfl_xor(rm, 16, 32);
    const float oZ = __shfl_xor(Z,  16, 32);
    const float oW = __shfl_xor(W,  16, 32);
    const float nm = fmaxf(rm, om);
    const float se = __builtin_amdgcn_exp2f(rm - nm);
    const float so = __builtin_amdgcn_exp2f(om - nm);
    const float Zt = fmaf(Z, se, oZ * so);
    const float Wt = fmaf(W, se, oW * so);

    if (!hi) outp[n] = Wt / Zt;   // A_t (dir=0) or V_s (dir=1)
}

// ---------------------------------------------------------------------------
// Kernel 3: final linear + literal (degenerate, size-1 axis) LayerNorm + ReLU.
// ---------------------------------------------------------------------------
__global__ __launch_bounds__(256)
void cf_final(const float* __restrict__ xa, const float* __restrict__ xv,
              const float* __restrict__ ws, float* __restrict__ out)
{
    const int i = blockIdx.x * blockDim.x + threadIdx.x;
    if (i >= BATCH * SEQ) return;

    const float c6 = ws[4], c7 = ws[5], c8 = ws[6], g = ws[7], be = ws[8];
    const float A = ws[WS_A_OFF + i];
    const float V = ws[WS_V_OFF + i];

    const float y  = fmaf(c6, A + xa[i], fmaf(c7, V + xv[i], c8));
    const float mu = y;                     // mean over size-1 axis
    const float d0 = y - mu;                // exactly 0 for finite y (kept literal)
    const float var = d0 * d0;
    const float yn = fmaf(d0 * rsqrtf(var + EPSV), g, be);
    out[i] = fmaxf(yn, 0.f);
}

extern "C" void kernel_launch(void* const* d_in, const int* in_sizes, int n_in,
                              void* d_out, int out_size, void* d_ws, size_t ws_size,
                              hipStream_t stream) {
    const float* audio  = (const float*)d_in[0];   // [8,2048,1]
    const float* visual = (const float*)d_in[1];   // [8,2048,1]
    const float* Wa     = (const float*)d_in[2];   // [256,1]
    const float* ba     = (const float*)d_in[3];   // [256]
    const float* Wv     = (const float*)d_in[4];   // [256,1]
    const float* bv     = (const float*)d_in[5];   // [256]
    const float* corr   = (const float*)d_in[6];   // [256,256]
    const float* Wb     = (const float*)d_in[7];   // [1,512]
    const float* bb     = (const float*)d_in[8];   // [1]
    const float* gam    = (const float*)d_in[9];   // [1]
    const float* bet    = (const float*)d_in[10];  // [1]
    float* out = (float*)d_out;                    // [8,2048,1]
    float* ws  = (float*)d_ws;                     // needs (16 + 2*8*2048)*4 = ~128 KB

    cf_scalars<<<1, 256, 0, stream>>>(Wa, ba, Wv, bv, corr, Wb,
                                      bb, gam, bet, ws);

    // 8 batches x 16 query-chunks x 2 directions = 256 blocks; 8 waves/block
    cf_attention<<<BATCH * 16 * 2, 256, 0, stream>>>(audio, visual, ws);

    cf_final<<<(BATCH * SEQ + 255) / 256, 256, 0, stream>>>(audio, visual, ws, out);
}